// QuantumBottleneckLayer_14293651161639
// MI455X (gfx1250) — compile-verified
//
#include <hip/hip_runtime.h>
#include <math.h>

typedef __attribute__((ext_vector_type(2))) float v2f;
typedef __attribute__((ext_vector_type(8))) float v8f;

#define B_ROWS 16384
#define D_DIM  768
#define H_DIM  256
#define NQ     8
#define TM     16

// padded LDS strides (avoid 64-bank conflicts; 768%64==0, 256%64==0)
#define SX_LD (D_DIM + 4)
#define SH_LD (H_DIM + 4)
#define SE_LD (H_DIM + 4)
#define SY_LD (D_DIM + 4)

__device__ __forceinline__ float gelu_exact(float v) {
  return 0.5f * v * (1.0f + erff(v * 0.70710678118654752f));
}

#define WMMA_F32(a, b, c) \
  __builtin_amdgcn_wmma_f32_16x16x4_f32(false, (a), false, (b), (short)0, (c), false, false)

// ---------------------------------------------------------------------------
// Kernel 1: h = gelu(x@W1+b1); feats = h@W2+b2   (fused, h stays in LDS)
// grid = 1024 blocks x 128 threads (4 waves); block owns 16 rows.
// Each wave keeps 4 accumulators (its 4 N-tiles) so the A fragment is loaded
// from LDS once per K-step and fed to 4 back-to-back WMMAs.
// ---------------------------------------------------------------------------
__global__ __launch_bounds__(128)
void k_in_mlp(const float* __restrict__ x, const float* __restrict__ W1,
              const float* __restrict__ bs1, const float* __restrict__ W2,
              const float* __restrict__ bs2, float* __restrict__ feats) {
  __shared__ __align__(16) float sx[TM * SX_LD];
  __shared__ float sh[TM * SH_LD];
  const int tid  = threadIdx.x;
  const int lane = tid & 31;
  const int wv   = tid >> 5;
  const int m0   = blockIdx.x * TM;

  // stage x tile (16 x 768) into LDS, coalesced float4 loads
  {
    const float4* xg = (const float4*)(x + (size_t)m0 * D_DIM);
    for (int i = tid; i < TM * (D_DIM / 4); i += 128) {
      int m = i / (D_DIM / 4);
      int c = i - m * (D_DIM / 4);
      *(float4*)(sx + m * SX_LD + 4 * c) = xg[(size_t)m * (D_DIM / 4) + c];
    }
  }
  __syncthreads();

  const int half = lane >> 4;   // wmma fragment half (K split)
  const int l15  = lane & 15;   // M index for A, N index for B/C
  const int n0   = wv * 64;     // this wave's 4 N-tiles: n0, n0+16, n0+32, n0+48

  v8f acc0 = {}, acc1 = {}, acc2 = {}, acc3 = {};
  const float* Ar = sx + l15 * SX_LD + 2 * half;                 // A[m][k+2*half+v]
  const float* Bp = W1 + (size_t)(2 * half) * H_DIM + n0 + l15;  // B[k+2*half+v][n]
#pragma unroll 2
  for (int k = 0; k < D_DIM; k += 4) {
    const float* B0 = Bp + (size_t)k * H_DIM;
    const float* B1 = B0 + H_DIM;
    __builtin_prefetch(B0 + 16 * H_DIM, 0, 0);   // stream W1 rows ahead (L2)
    v2f a;
    a[0] = Ar[k];
    a[1] = Ar[k + 1];
    v2f b0, b1, b2, b3;
    b0[0] = B0[0];  b0[1] = B1[0];
    b1[0] = B0[16]; b1[1] = B1[16];
    b2[0] = B0[32]; b2[1] = B1[32];
    b3[0] = B0[48]; b3[1] = B1[48];
    acc0 = WMMA_F32(a, b0, acc0);
    acc1 = WMMA_F32(a, b1, acc1);
    acc2 = WMMA_F32(a, b2, acc2);
    acc3 = WMMA_F32(a, b3, acc3);
  }
#pragma unroll
  for (int r = 0; r < 8; ++r) {
    int m = r + 8 * half;
    int n = n0 + l15;
    sh[m * SH_LD + n +  0] = gelu_exact(acc0[r] + bs1[n +  0]);
    sh[m * SH_LD + n + 16] = gelu_exact(acc1[r] + bs1[n + 16]);
    sh[m * SH_LD + n + 32] = gelu_exact(acc2[r] + bs1[n + 32]);
    sh[m * SH_LD + n + 48] = gelu_exact(acc3[r] + bs1[n + 48]);
  }
  __syncthreads();

  // feats (16x8): one output per thread, 256-length dot from LDS
  {
    int m = tid >> 3, j = tid & 7;
    const float* hr = sh + m * SH_LD;
    float s = bs2[j];
#pragma unroll 8
    for (int k = 0; k < H_DIM; ++k) s += hr[k] * W2[k * NQ + j];
    feats[(size_t)(m0 + m) * NQ + j] = s;
  }
}

// ---------------------------------------------------------------------------
// Kernel 2: 8-qubit statevector sim, one wave per sample.
// amp index a = lane*8 + r ; wire i <-> bit (7-i); lane carries bits 7..3.
// ---------------------------------------------------------------------------
__device__ __forceinline__ float2 cmul(float2 a, float2 b) {
  return make_float2(a.x * b.x - a.y * b.y, a.x * b.y + a.y * b.x);
}
__device__ __forceinline__ float2 cadd(float2 a, float2 b) {
  return make_float2(a.x + b.x, a.y + b.y);
}

__device__ __forceinline__ void apply1q(float2* amp, float2 U00, float2 U01,
                                        float2 U10, float2 U11, int bit, int lane) {
  if (bit < 3) {                       // local bit: pair within the 8 regs
    const int tm = 1 << bit;
#pragma unroll
    for (int r = 0; r < 8; ++r) {
      if (!(r & tm)) {
        float2 a0 = amp[r], a1 = amp[r | tm];
        amp[r]      = cadd(cmul(U00, a0), cmul(U01, a1));
        amp[r | tm] = cadd(cmul(U10, a0), cmul(U11, a1));
      }
    }
  } else {                             // lane bit: cross-lane via shfl_xor (wave32)
    const int lm = 1 << (bit - 3);
    const bool hi = (lane & lm) != 0;
#pragma unroll
    for (int r = 0; r < 8; ++r) {
      float2 self = amp[r], part;
      part.x = __shfl_xor(self.x, lm, 32);
      part.y = __shfl_xor(self.y, lm, 32);
      float2 a0 = hi ? part : self;
      float2 a1 = hi ? self : part;
      float2 lo  = cadd(cmul(U00, a0), cmul(U01, a1));
      float2 hiv = cadd(cmul(U10, a0), cmul(U11, a1));
      amp[r] = hi ? hiv : lo;
    }
  }
}

__device__ __forceinline__ void cnotg(float2* amp, int cbit, int tbit, int lane) {
  const int cml = (cbit >= 3) ? (1 << (cbit - 3)) : 0;  // control in lane bits
  const int cmr = (cbit < 3) ? (1 << cbit) : 0;         // control in reg bits
  if (tbit < 3) {
    const int tm = 1 << tbit;
#pragma unroll
    for (int r = 0; r < 8; ++r) {
      if (!(r & tm)) {
        bool ctrl = cml ? ((lane & cml) != 0) : ((r & cmr) != 0);
        float2 a0 = amp[r], a1 = amp[r | tm];
        amp[r]      = ctrl ? a1 : a0;
        amp[r | tm] = ctrl ? a0 : a1;
      }
    }
  } else {
    const int lm = 1 << (tbit - 3);
#pragma unroll
    for (int r = 0; r < 8; ++r) {
      float2 self = amp[r], part;
      part.x = __shfl_xor(self.x, lm, 32);
      part.y = __shfl_xor(self.y, lm, 32);
      bool ctrl = cml ? ((lane & cml) != 0) : ((r & cmr) != 0);
      amp[r] = ctrl ? part : self;
    }
  }
}

__global__ __launch_bounds__(256)
void k_quantum(const float* __restrict__ feats, const float* __restrict__ qw,
               float* __restrict__ zout) {
  const int lane   = threadIdx.x & 31;
  const int sample = blockIdx.x * (256 >> 5) + (threadIdx.x >> 5);

  float2 amp[8];
#pragma unroll
  for (int r = 0; r < 8; ++r) amp[r] = make_float2(0.f, 0.f);
  amp[0].x = (lane == 0) ? 1.0f : 0.0f;   // |0...0>

  float f[NQ];
#pragma unroll
  for (int i = 0; i < NQ; ++i) f[i] = feats[(size_t)sample * NQ + i];

#pragma unroll
  for (int l = 0; l < 2; ++l) {
#pragma unroll
    for (int w = 0; w < NQ; ++w) {
      const float w0 = qw[(l * NQ + w) * 3 + 0];
      const float w1 = qw[(l * NQ + w) * 3 + 1];
      const float w2 = qw[(l * NQ + w) * 3 + 2];
      float sb, cb; sincosf(0.5f * w1, &sb, &cb);
      float sp, cp; sincosf(0.5f * (w0 + w2), &sp, &cp);  // e^{-i(w0+w2)/2}=cp-isp
      float sm, cm; sincosf(0.5f * (w0 - w2), &sm, &cm);  // e^{+i(w0-w2)/2}=cm+ism
      float2 C00 = make_float2( cb * cp, -cb * sp);
      float2 C11 = make_float2( cb * cp,  cb * sp);
      float2 C01 = make_float2(-sb * cm, -sb * sm);
      float2 C10 = make_float2( sb * cm, -sb * sm);
      float sf, cf; sincosf(0.5f * f[w], &sf, &cf);       // fold RY(f_w): U = C @ RY
      float2 U00 = make_float2(C00.x * cf + C01.x * sf, C00.y * cf + C01.y * sf);
      float2 U01 = make_float2(C01.x * cf - C00.x * sf, C01.y * cf - C00.y * sf);
      float2 U10 = make_float2(C10.x * cf + C11.x * sf, C10.y * cf + C11.y * sf);
      float2 U11 = make_float2(C11.x * cf - C10.x * sf, C11.y * cf - C10.y * sf);
      apply1q(amp, U00, U01, U10, U11, 7 - w, lane);
    }
#pragma unroll
    for (int i = 0; i < NQ; ++i) cnotg(amp, 7 - i, 7 - ((i + 1) & 7), lane);
  }

  // <Z_w> = sum_a |amp[a]|^2 * (1 - 2*bit(a, 7-w))
  float zp[NQ];
#pragma unroll
  for (int w = 0; w < NQ; ++w) zp[w] = 0.f;
#pragma unroll
  for (int r = 0; r < 8; ++r) {
    float p = amp[r].x * amp[r].x + amp[r].y * amp[r].y;
    int a = (lane << 3) | r;
#pragma unroll
    for (int w = 0; w < NQ; ++w) zp[w] += ((a >> (7 - w)) & 1) ? -p : p;
  }
#pragma unroll
  for (int off = 16; off; off >>= 1) {
#pragma unroll
    for (int w = 0; w < NQ; ++w) zp[w] += __shfl_xor(zp[w], off, 32);
  }
  if (lane == 0) {
#pragma unroll
    for (int w = 0; w < NQ; ++w) zout[(size_t)sample * NQ + w] = zp[w];
  }
}

// ---------------------------------------------------------------------------
// Kernel 3: e = gelu(z@W3+b3); y = x + e@W4 + b4; out = LayerNorm(y)
// grid = 1024 blocks x 128 threads (4 waves); block owns 16 rows.
// Each wave owns 12 N-tiles, processed as 3 groups of 4 concurrent
// accumulators so the A fragment is loaded once per K-step.
// ---------------------------------------------------------------------------
__global__ __launch_bounds__(128)
void k_out_mlp(const float* __restrict__ z, const float* __restrict__ W3,
               const float* __restrict__ bs3, const float* __restrict__ W4,
               const float* __restrict__ bs4, const float* __restrict__ x,
               const float* __restrict__ gamma, const float* __restrict__ beta,
               float* __restrict__ out) {
  __shared__ float se[TM * SE_LD];        // e tile 16x256
  __shared__ float sy[TM * SY_LD];        // y tile 16x768
  __shared__ float sred[TM * 8 * 2];
  __shared__ float smu[TM], srstd[TM];
  const int tid  = threadIdx.x;
  const int lane = tid & 31;
  const int wv   = tid >> 5;
  const int m0   = blockIdx.x * TM;

  // phase 1: e[m][k] = gelu(dot8(z[m], W3[:,k]) + b3[k])
  for (int i = tid; i < TM * H_DIM; i += 128) {
    int m = i >> 8;
    int k = i & (H_DIM - 1);
    const float* zr = z + (size_t)(m0 + m) * NQ;
    float s = bs3[k];
#pragma unroll
    for (int j = 0; j < NQ; ++j) s += zr[j] * W3[j * H_DIM + k];
    se[m * SE_LD + k] = gelu_exact(s);
  }
  __syncthreads();

  const int half = lane >> 4;
  const int l15  = lane & 15;

  // phase 2: 3 groups of 4 concurrent (16x16) tiles per wave
  for (int g = 0; g < 3; ++g) {
    const int n0 = wv * 192 + g * 64;
    v8f acc0 = {}, acc1 = {}, acc2 = {}, acc3 = {};
    const float* Ar = se + l15 * SE_LD + 2 * half;
    const float* Bp = W4 + (size_t)(2 * half) * D_DIM + n0 + l15;
#pragma unroll 2
    for (int k = 0; k < H_DIM; k += 4) {
      const float* B0 = Bp + (size_t)k * D_DIM;
      const float* B1 = B0 + D_DIM;
      __builtin_prefetch(B0 + 16 * D_DIM, 0, 0);   // stream W4 rows ahead (L2)
      v2f a;
      a[0] = Ar[k];
      a[1] = Ar[k + 1];
      v2f b0, b1, b2, b3;
      b0[0] = B0[0];  b0[1] = B1[0];
      b1[0] = B0[16]; b1[1] = B1[16];
      b2[0] = B0[32]; b2[1] = B1[32];
      b3[0] = B0[48]; b3[1] = B1[48];
      acc0 = WMMA_F32(a, b0, acc0);
      acc1 = WMMA_F32(a, b1, acc1);
      acc2 = WMMA_F32(a, b2, acc2);
      acc3 = WMMA_F32(a, b3, acc3);
    }
#pragma unroll
    for (int r = 0; r < 8; ++r) {
      int m = r + 8 * half;
      int n = n0 + l15;
      const float* xr = x + (size_t)(m0 + m) * D_DIM;
      sy[m * SY_LD + n +  0] = acc0[r] + bs4[n +  0] + xr[n +  0];
      sy[m * SY_LD + n + 16] = acc1[r] + bs4[n + 16] + xr[n + 16];
      sy[m * SY_LD + n + 32] = acc2[r] + bs4[n + 32] + xr[n + 32];
      sy[m * SY_LD + n + 48] = acc3[r] + bs4[n + 48] + xr[n + 48];
    }
  }
  __syncthreads();

  // phase 3: LayerNorm. 8 threads per row, 96 columns each.
  {
    int m = tid >> 3, part = tid & 7;
    const float* yr = sy + m * SY_LD + part * 96;
    float s = 0.f, s2 = 0.f;
    for (int c = 0; c < 96; ++c) { float v = yr[c]; s += v; s2 += v * v; }
    sred[(m * 8 + part) * 2 + 0] = s;
    sred[(m * 8 + part) * 2 + 1] = s2;
  }
  __syncthreads();
  if (tid < TM) {
    float s = 0.f, s2 = 0.f;
    for (int p = 0; p < 8; ++p) {
      s  += sred[(tid * 8 + p) * 2 + 0];
      s2 += sred[(tid * 8 + p) * 2 + 1];
    }
    float mu  = s * (1.0f / D_DIM);
    float var = s2 * (1.0f / D_DIM) - mu * mu;
    smu[tid]   = mu;
    srstd[tid] = rsqrtf(var + 1e-5f);
  }
  __syncthreads();
  for (int i = tid; i < TM * D_DIM; i += 128) {
    int m = i / D_DIM;
    int c = i - m * D_DIM;
    float v = (sy[m * SY_LD + c] - smu[m]) * srstd[m];
    out[(size_t)(m0 + m) * D_DIM + c] = gamma[c] * v + beta[c];
  }
}

// ---------------------------------------------------------------------------
extern "C" void kernel_launch(void* const* d_in, const int* in_sizes, int n_in,
                              void* d_out, int out_size, void* d_ws, size_t ws_size,
                              hipStream_t stream) {
  (void)in_sizes; (void)n_in; (void)out_size; (void)ws_size;
  const float* x     = (const float*)d_in[0];
  const float* W1    = (const float*)d_in[1];
  const float* b1    = (const float*)d_in[2];
  const float* W2    = (const float*)d_in[3];
  const float* b2    = (const float*)d_in[4];
  const float* qw    = (const float*)d_in[5];
  const float* W3    = (const float*)d_in[6];
  const float* b3    = (const float*)d_in[7];
  const float* W4    = (const float*)d_in[8];
  const float* b4    = (const float*)d_in[9];
  const float* gamma = (const float*)d_in[10];
  const float* beta  = (const float*)d_in[11];
  float* outp = (float*)d_out;

  float* feats = (float*)d_ws;                       // 16384*8 f32
  float* zvals = feats + (size_t)B_ROWS * NQ;        // 16384*8 f32

  k_in_mlp<<<B_ROWS / TM, 128, 0, stream>>>(x, W1, b1, W2, b2, feats);
  k_quantum<<<B_ROWS / 8, 256, 0, stream>>>(feats, qw, zvals);
  k_out_mlp<<<B_ROWS / TM, 128, 0, stream>>>(zvals, W3, b3, W4, b4, x,
                                             gamma, beta, outp);
}